// ContractiveREN_54288386622001
// MI455X (gfx1250) — compile-verified
//
// ContractiveREN on MI455X (gfx1250, CDNA5, wave32).
//
// Strategy (latency-bound sequential scan, ~53 GFLOP total):
//  * One-time derivation: H = X^T X + eps*I via V_WMMA_F32_16X16X4_F32,
//    Gauss-Jordan inverses for E (128x128) and C2*C2^T (32x32) in LDS,
//    E^-1 folded into weights (Ftil=E^-1 F, B1til=E^-1 B1, B2til=E^-1 B2).
//  * Persistent scan kernel: 16 blocks (one 16-row WMMA batch tile each),
//    256 threads = 8 waves. ALL weights (~320KB) live in LDS (staged with
//    GLOBAL_LOAD_ASYNC_TO_LDS, ASYNCcnt-tracked), bank-conflict-padded
//    strides. Every matmul is f32 WMMA (full precision through the 1024-step
//    tanh recurrence).
//  * Inner q=128 forward substitution: each thread owns 8 CONSECUTIVE
//    channels, so each 8-block is resolved locally by its owner (triangular
//    in-register updates) => 16 barriers per timestep instead of 128. The
//    only remaining serial cost is the irreducible 128-long tanh chain,
//    shortened via v_exp_f32-based tanh and reciprocal-Lambda.
//
// Workspace use: ~1.25 MB of d_ws (floats).

#include <hip/hip_runtime.h>
#include <hip/hip_bf16.h>
#include <math.h>

typedef float v2f __attribute__((ext_vector_type(2)));
typedef float v8f __attribute__((ext_vector_type(8)));

// ---------------------------------------------------------------------------
// WMMA f32 16x16x4 helper.  D = A(16x4) * B(4x16) + C, striped across wave32.
// A layout: lane l -> m = l&15 ; VGPR0: K = (l>=16?2:0), VGPR1: K+1
// B layout: lane l -> n = l&15 ; VGPR0: K = (l>=16?2:0), VGPR1: K+1
// C/D layout: lane l -> n = l&15, VGPR v -> m = v + 8*(l>=16)
// ---------------------------------------------------------------------------
__device__ __forceinline__ v8f wmma4(v2f a, v2f b, v8f c) {
  return __builtin_amdgcn_wmma_f32_16x16x4_f32(
      /*neg_a=*/false, a, /*neg_b=*/false, b,
      /*c_mod=*/(short)0, c, /*reuse_a=*/false, /*reuse_b=*/false);
}

#define ZERO8 {0.f,0.f,0.f,0.f,0.f,0.f,0.f,0.f}

// fast tanh on the serial critical path: v_exp_f32 + v_rcp_f32
__device__ __forceinline__ float fast_tanh(float x) {
  float e = __expf(2.0f * x);
  return 1.0f - 2.0f * __builtin_amdgcn_rcpf(e + 1.0f);
}

// ---------------------------------------------------------------------------
// Async global->LDS staging (CDNA5 ASYNCcnt path), with scalar fallback.
// Probe-discovered signature: (int4 as1*, int4 as3*, imm offset, imm cpol).
// ---------------------------------------------------------------------------
#if defined(__HIP_DEVICE_COMPILE__) &&                                   \
    __has_builtin(__builtin_amdgcn_global_load_async_to_lds_b128) &&     \
    __has_builtin(__builtin_amdgcn_global_load_async_to_lds_b64)
#define HAVE_ASYNC_LDS 1
typedef int gv4i __attribute__((vector_size(16)));
typedef int gv2i __attribute__((vector_size(8)));
typedef __attribute__((address_space(1))) gv4i* g4p;   // global int4*
typedef __attribute__((address_space(3))) gv4i* l4p;   // LDS int4*
typedef __attribute__((address_space(1))) gv2i* g2p;   // global int2*
typedef __attribute__((address_space(3))) gv2i* l2p;   // LDS int2*
__device__ __forceinline__ void async_cp16(const float* g, float* l) {
  __builtin_amdgcn_global_load_async_to_lds_b128((g4p)g, (l4p)l, 0, 0);
}
__device__ __forceinline__ void async_cp8(const float* g, float* l) {
  __builtin_amdgcn_global_load_async_to_lds_b64((g2p)g, (l2p)l, 0, 0);
}
__device__ __forceinline__ void wait_async0() {
#if __has_builtin(__builtin_amdgcn_s_wait_asynccnt)
  __builtin_amdgcn_s_wait_asynccnt(0);
#else
  asm volatile("s_wait_asynccnt 0x0" ::: "memory");
#endif
}
#endif

// ===========================================================================
// H = X^T X + EPS*I   (X: 384x384, H: 384x384).  grid (24,24) x 32 threads.
// ===========================================================================
__global__ void k_gemm_AtA(const float* __restrict__ X, float* __restrict__ H) {
  const int lane = threadIdx.x & 31, half = lane >> 4, lm = lane & 15;
  const int kb = half << 1;
  const int i0 = blockIdx.y * 16, j0 = blockIdx.x * 16;
  v8f acc = ZERO8;
  for (int k0 = 0; k0 < 384; k0 += 4) {
    v2f a = { X[(k0 + kb) * 384 + i0 + lm], X[(k0 + kb + 1) * 384 + i0 + lm] };
    v2f b = { X[(k0 + kb) * 384 + j0 + lm], X[(k0 + kb + 1) * 384 + j0 + lm] };
    acc = wmma4(a, b, acc);
  }
#pragma unroll
  for (int v = 0; v < 8; ++v) {
    int m = v + half * 8;
    float val = acc[v];
    if (i0 + m == j0 + lm) val += 0.001f;   // EPS on diagonal
    H[(i0 + m) * 384 + j0 + lm] = val;
  }
}

// ===========================================================================
// Generic C[M,N] = A[M,K] @ B[K,N], all row-major. grid (N/16, M/16) x 32.
// ===========================================================================
__global__ void k_gemm_rrr(const float* __restrict__ A, int lda,
                           const float* __restrict__ B, int ldb,
                           float* __restrict__ C, int ldc, int K) {
  const int lane = threadIdx.x & 31, half = lane >> 4, lm = lane & 15;
  const int kb = half << 1;
  const int m0 = blockIdx.y * 16, n0 = blockIdx.x * 16;
  v8f acc = ZERO8;
  for (int k0 = 0; k0 < K; k0 += 4) {
    v2f a = { A[(m0 + lm) * lda + k0 + kb], A[(m0 + lm) * lda + k0 + kb + 1] };
    v2f b = { B[(k0 + kb) * ldb + n0 + lm], B[(k0 + kb + 1) * ldb + n0 + lm] };
    acc = wmma4(a, b, acc);
  }
#pragma unroll
  for (int v = 0; v < 8; ++v)
    C[(m0 + v + half * 8) * ldc + n0 + lm] = acc[v];
}

// ===========================================================================
// Generic C[M,N] = A[M,K] @ W^T  (W row-major NxK). grid (N/16, M/16) x 32.
// ===========================================================================
__global__ void k_gemm_rnt(const float* __restrict__ A, int lda,
                           const float* __restrict__ W, int ldw,
                           float* __restrict__ C, int ldc, int K) {
  const int lane = threadIdx.x & 31, half = lane >> 4, lm = lane & 15;
  const int kb = half << 1;
  const int m0 = blockIdx.y * 16, n0 = blockIdx.x * 16;
  v8f acc = ZERO8;
  for (int k0 = 0; k0 < K; k0 += 4) {
    v2f a = { A[(m0 + lm) * lda + k0 + kb], A[(m0 + lm) * lda + k0 + kb + 1] };
    v2f b = { W[(n0 + lm) * ldw + k0 + kb], W[(n0 + lm) * ldw + k0 + kb + 1] };
    acc = wmma4(a, b, acc);
  }
#pragma unroll
  for (int v = 0; v < 8; ++v)
    C[(m0 + v + half * 8) * ldc + n0 + lm] = acc[v];
}

// ===========================================================================
// Derive E, C1, F_, B1, packed D11, Lam from H (384x384) and Y (128x128).
// ===========================================================================
__global__ void k_derive(const float* __restrict__ H, const float* __restrict__ Y,
                         float* __restrict__ E, float* __restrict__ C1,
                         float* __restrict__ F_, float* __restrict__ B1,
                         float* __restrict__ D11p, float* __restrict__ Lam) {
  int idx = blockIdx.x * blockDim.x + threadIdx.x;
  if (idx >= 128 * 128) return;
  const int i = idx >> 7, j = idx & 127;
  E[idx]  = 0.5f * (H[i * 384 + j] + H[(256 + i) * 384 + 256 + j]
                    + Y[idx] - Y[j * 128 + i]);       // ALPHA = 1
  C1[idx] = -H[(128 + i) * 384 + j];                  // C1 = -H21
  F_[idx] =  H[(256 + i) * 384 + j];                  // F  =  H31
  B1[idx] =  H[(256 + i) * 384 + 128 + j];            // B1 =  H32
  if (j < i)  D11p[(i * (i - 1)) / 2 + j] = -H[(128 + i) * 384 + 128 + j];
  if (j == i) Lam[i] = 0.5f * H[(128 + i) * 384 + 128 + i];
}

// ===========================================================================
// Gauss-Jordan inverse of 128x128 (single block, 256 threads, dynamic LDS).
// ===========================================================================
extern __shared__ float smem[];

__global__ void k_invert128(const float* __restrict__ E, float* __restrict__ Einv) {
  float* aug = smem;                 // 128 x 257 (256 cols used, padded)
  float* fac = smem + 128 * 257;     // 128
  float* piv = fac + 128;            // 1
  const int tid = threadIdx.x;
  for (int idx = tid; idx < 128 * 256; idx += 256) {
    int r = idx >> 8, c = idx & 255;
    aug[r * 257 + c] = (c < 128) ? E[r * 128 + c] : ((c - 128) == r ? 1.f : 0.f);
  }
  __syncthreads();
  for (int p = 0; p < 128; ++p) {
    if (tid == 0) piv[0] = 1.0f / aug[p * 257 + p];
    __syncthreads();
    aug[p * 257 + tid] *= piv[0];              // normalize pivot row (256 cols)
    __syncthreads();
    if (tid < 128) fac[tid] = (tid == p) ? 0.f : aug[tid * 257 + p];
    __syncthreads();
    {
      const int r = tid & 127, h = tid >> 7;   // each thread: 1 row, 128 cols
      const float f = fac[r];
      const float* prow = &aug[p * 257 + h * 128];
      float* rrow = &aug[r * 257 + h * 128];
#pragma unroll 4
      for (int c = 0; c < 128; ++c) rrow[c] -= f * prow[c];
    }
    __syncthreads();
  }
  for (int idx = tid; idx < 128 * 128; idx += 256) {
    int r = idx >> 7, c = idx & 127;
    Einv[idx] = aug[r * 257 + 128 + c];
  }
}

// ===========================================================================
// pinv path: G = (C2 C2^T)^-1 C2  (32x128).  One block, 256 threads.
// ===========================================================================
__global__ void k_small32(const float* __restrict__ C2, float* __restrict__ G) {
  __shared__ float aug[32 * 65];
  __shared__ float fac[32];
  __shared__ float piv;
  const int tid = threadIdx.x;
  for (int idx = tid; idx < 1024; idx += 256) {
    int i = idx >> 5, j = idx & 31;
    float s = 0.f;
    for (int k = 0; k < 128; ++k) s += C2[i * 128 + k] * C2[j * 128 + k];
    aug[i * 65 + j] = s;
    aug[i * 65 + 32 + j] = (i == j) ? 1.f : 0.f;
  }
  __syncthreads();
  for (int p = 0; p < 32; ++p) {
    if (tid == 0) piv = 1.0f / aug[p * 65 + p];
    __syncthreads();
    if (tid < 64) aug[p * 65 + tid] *= piv;
    __syncthreads();
    if (tid < 32) fac[tid] = (tid == p) ? 0.f : aug[tid * 65 + p];
    __syncthreads();
    for (int idx = tid; idx < 32 * 64; idx += 256) {
      int r = idx >> 6, c = idx & 63;
      aug[r * 65 + c] -= fac[r] * aug[p * 65 + c];
    }
    __syncthreads();
  }
  for (int idx = tid; idx < 32 * 128; idx += 256) {
    int i = idx >> 7, j = idx & 127;
    float s = 0.f;
    for (int k = 0; k < 32; ++k) s += aug[i * 65 + 32 + k] * C2[k * 128 + j];
    G[idx] = s;
  }
}

// ===========================================================================
// Persistent scan kernel.  grid = B/16 blocks, 256 threads (8 waves).
// LDS (floats, padded strides to dodge 16-way bank conflicts):
//   sC1 128x132 | sFt 128x132 | sB1t 128x132 | sD12 128x36 | sB2t 128x36
//   sC2 32x130  | sD21 32x130 | sD22 32x36   | sD11 packed-tri 8128
//   sLam 128 (stores 1/Lam) | sX 16x132 | sW 16x132  => 327,424 B (<=320KB)
// ===========================================================================
#define SCAN_LDS_FLOATS (3*128*132 + 2*128*36 + 2*32*130 + 32*36 + 8128 + 128 + 2*16*132)

__global__ __launch_bounds__(256)
void k_scan(const float* __restrict__ u,
            const float* __restrict__ gC1, const float* __restrict__ gFt,
            const float* __restrict__ gB1t, const float* __restrict__ gD11p,
            const float* __restrict__ gLam, const float* __restrict__ gB2t,
            const float* __restrict__ gD12, const float* __restrict__ gC2,
            const float* __restrict__ gD21, const float* __restrict__ gD22,
            const float* __restrict__ x0, float* __restrict__ out, int T) {
  float* sC1  = smem;
  float* sFt  = sC1 + 128 * 132;
  float* sB1t = sFt + 128 * 132;
  float* sD12 = sB1t + 128 * 132;
  float* sB2t = sD12 + 128 * 36;
  float* sC2  = sB2t + 128 * 36;
  float* sD21 = sC2 + 32 * 130;
  float* sD22 = sD21 + 32 * 130;
  float* sD11 = sD22 + 32 * 36;
  float* sLam = sD11 + 8128;           // holds 1/Lam
  float* sX   = sLam + 128;
  float* sW   = sX + 16 * 132;

  const int tid = threadIdx.x;
  const int b0 = blockIdx.x * 16;

  // ---- stage weights into LDS (one-time) ----
  if (tid < 128) sLam[tid] = 1.0f / gLam[tid];   // reciprocal for the scan
#ifdef HAVE_ASYNC_LDS
  for (int idx = tid; idx < 128 * 32; idx += 256) {
    int rr = idx >> 5, c4 = (idx & 31) * 4;
    async_cp16(gC1 + rr * 128 + c4, sC1 + rr * 132 + c4);
    async_cp16(gFt + rr * 128 + c4, sFt + rr * 132 + c4);
    async_cp16(gB1t + rr * 128 + c4, sB1t + rr * 132 + c4);
  }
  for (int idx = tid; idx < 128 * 8; idx += 256) {
    int rr = idx >> 3, c4 = (idx & 7) * 4;
    async_cp16(gD12 + rr * 32 + c4, sD12 + rr * 36 + c4);
    async_cp16(gB2t + rr * 32 + c4, sB2t + rr * 36 + c4);
  }
  for (int idx = tid; idx < 32 * 64; idx += 256) {   // stride 130 rows: 8B align
    int rr = idx >> 6, c2 = (idx & 63) * 2;
    async_cp8(gC2 + rr * 128 + c2, sC2 + rr * 130 + c2);
    async_cp8(gD21 + rr * 128 + c2, sD21 + rr * 130 + c2);
  }
  for (int idx = tid; idx < 32 * 8; idx += 256) {
    int rr = idx >> 3, c4 = (idx & 7) * 4;
    async_cp16(gD22 + rr * 32 + c4, sD22 + rr * 36 + c4);
  }
  for (int idx = tid; idx < 8128 / 4; idx += 256)
    async_cp16(gD11p + idx * 4, sD11 + idx * 4);
  for (int idx = tid; idx < 16 * 32; idx += 256) {
    int m = idx >> 5, c4 = (idx & 31) * 4;
    async_cp16(x0 + (b0 + m) * 128 + c4, sX + m * 132 + c4);
  }
  wait_async0();
#else
  for (int idx = tid; idx < 128 * 128; idx += 256) {
    int rr = idx >> 7, c = idx & 127;
    sC1[rr * 132 + c]  = gC1[idx];
    sFt[rr * 132 + c]  = gFt[idx];
    sB1t[rr * 132 + c] = gB1t[idx];
  }
  for (int idx = tid; idx < 128 * 32; idx += 256) {
    int rr = idx >> 5, c = idx & 31;
    sD12[rr * 36 + c] = gD12[idx];
    sB2t[rr * 36 + c] = gB2t[idx];
  }
  for (int idx = tid; idx < 32 * 128; idx += 256) {
    int rr = idx >> 7, c = idx & 127;
    sC2[rr * 130 + c]  = gC2[idx];
    sD21[rr * 130 + c] = gD21[idx];
  }
  for (int idx = tid; idx < 32 * 32; idx += 256)
    sD22[(idx >> 5) * 36 + (idx & 31)] = gD22[idx];
  for (int idx = tid; idx < 8128; idx += 256) sD11[idx] = gD11p[idx];
  for (int idx = tid; idx < 16 * 128; idx += 256) {
    int m = idx >> 7, k = idx & 127;
    sX[m * 132 + k] = x0[(b0 + m) * 128 + k];
  }
#endif
  __syncthreads();

  // wave / lane decomposition (WMMA fragment addressing)
  const int wid = tid >> 5;               // wave 0..7 (uniform per wave)
  const int lane = tid & 31, half = lane >> 4, lm = lane & 15;
  const int kb = half << 1;
  const int n0 = wid * 16;                // this wave's N-tile
  // scan-thread decomposition: row r, 8 owned consecutive i in [i0, i0+8)
  const int r = tid & 15, g = tid >> 4, i0 = g * 8;
  int ibase[8];
#pragma unroll
  for (int q = 0; q < 8; ++q) { int i = i0 + q; ibase[q] = (i * (i - 1)) / 2; }

  const float* uRow = u + (size_t)(b0 + lm) * (size_t)T * 32;  // A-fragment row

  for (int t = 0; t < T - 1; ++t) {
    const float* uA = uRow + t * 32;
    __builtin_prefetch(uA + 32, 0, 0);          // global_prefetch next step's u

    // ---- stage 1: a = x @ C1^T + u @ D12^T  -> sW (as staging for 'a') ----
    v8f acc = ZERO8;
#pragma unroll 8
    for (int k0 = 0; k0 < 128; k0 += 4) {
      v2f a = { sX[lm * 132 + k0 + kb], sX[lm * 132 + k0 + kb + 1] };
      v2f b = { sC1[(n0 + lm) * 132 + k0 + kb], sC1[(n0 + lm) * 132 + k0 + kb + 1] };
      acc = wmma4(a, b, acc);
    }
#pragma unroll
    for (int k0 = 0; k0 < 32; k0 += 4) {
      v2f a = { uA[k0 + kb], uA[k0 + kb + 1] };
      v2f b = { sD12[(n0 + lm) * 36 + k0 + kb], sD12[(n0 + lm) * 36 + k0 + kb + 1] };
      acc = wmma4(a, b, acc);
    }
#pragma unroll
    for (int v = 0; v < 8; ++v) sW[(v + half * 8) * 132 + n0 + lm] = acc[v];
    __syncthreads();

    // ---- stage 2: forward substitution w_i = tanh((a_i + D11 w)/Lam_i)
    //      8-wide rounds: owner resolves its block locally, 1 barrier/round.
    float av[8];
#pragma unroll
    for (int q = 0; q < 8; ++q) av[q] = sW[r * 132 + i0 + q];
    __syncthreads();                       // everyone captured 'a' before w_0
    for (int rd = 0; rd < 16; ++rd) {
      const int jb = rd * 8;
      if (g == rd) {                       // owner: serial 8-step resolve
#pragma unroll
        for (int q = 0; q < 8; ++q) {
          float w_ = fast_tanh(av[q] * sLam[jb + q]);
          sW[r * 132 + jb + q] = w_;
#pragma unroll
          for (int q2 = q + 1; q2 < 8; ++q2)
            av[q2] = fmaf(sD11[ibase[q2] + jb + q], w_, av[q2]);
        }
      }
      __syncthreads();                     // single barrier per 8 channels
      if (g > rd) {                        // downstream threads fold in block
#pragma unroll
        for (int jj = 0; jj < 8; ++jj) {
          const float wj = sW[r * 132 + jb + jj];
#pragma unroll
          for (int q = 0; q < 8; ++q)
            av[q] = fmaf(sD11[ibase[q] + jb + jj], wj, av[q]);
        }
      }
    }
    // last round's barrier already made sW (= w) visible to all waves

    // ---- stage 3: x_new = x @ Ftil^T + w @ B1til^T + u @ B2til^T ----
    v8f xn = ZERO8;
#pragma unroll 8
    for (int k0 = 0; k0 < 128; k0 += 4) {
      v2f a = { sX[lm * 132 + k0 + kb], sX[lm * 132 + k0 + kb + 1] };
      v2f b = { sFt[(n0 + lm) * 132 + k0 + kb], sFt[(n0 + lm) * 132 + k0 + kb + 1] };
      xn = wmma4(a, b, xn);
    }
#pragma unroll 8
    for (int k0 = 0; k0 < 128; k0 += 4) {
      v2f a = { sW[lm * 132 + k0 + kb], sW[lm * 132 + k0 + kb + 1] };
      v2f b = { sB1t[(n0 + lm) * 132 + k0 + kb], sB1t[(n0 + lm) * 132 + k0 + kb + 1] };
      xn = wmma4(a, b, xn);
    }
#pragma unroll
    for (int k0 = 0; k0 < 32; k0 += 4) {
      v2f a = { uA[k0 + kb], uA[k0 + kb + 1] };
      v2f b = { sB2t[(n0 + lm) * 36 + k0 + kb], sB2t[(n0 + lm) * 36 + k0 + kb + 1] };
      xn = wmma4(a, b, xn);
    }
    __syncthreads();                       // all reads of old sX complete
#pragma unroll
    for (int v = 0; v < 8; ++v) sX[(v + half * 8) * 132 + n0 + lm] = xn[v];
    __syncthreads();

    // ---- stage 4: y = x_new @ C2^T + w @ D21^T + u @ D22^T  (waves 0,1) ----
    if (wid < 2) {                         // wave-uniform => EXEC stays all-1
      v8f y = ZERO8;
#pragma unroll 8
      for (int k0 = 0; k0 < 128; k0 += 4) {
        v2f a = { sX[lm * 132 + k0 + kb], sX[lm * 132 + k0 + kb + 1] };
        v2f b = { sC2[(n0 + lm) * 130 + k0 + kb], sC2[(n0 + lm) * 130 + k0 + kb + 1] };
        y = wmma4(a, b, y);
      }
#pragma unroll 8
      for (int k0 = 0; k0 < 128; k0 += 4) {
        v2f a = { sW[lm * 132 + k0 + kb], sW[lm * 132 + k0 + kb + 1] };
        v2f b = { sD21[(n0 + lm) * 130 + k0 + kb], sD21[(n0 + lm) * 130 + k0 + kb + 1] };
        y = wmma4(a, b, y);
      }
#pragma unroll
      for (int k0 = 0; k0 < 32; k0 += 4) {
        v2f a = { uA[k0 + kb], uA[k0 + kb + 1] };
        v2f b = { sD22[(n0 + lm) * 36 + k0 + kb], sD22[(n0 + lm) * 36 + k0 + kb + 1] };
        y = wmma4(a, b, y);
      }
#pragma unroll
      for (int v = 0; v < 8; ++v)
        out[(size_t)(b0 + v + half * 8) * (size_t)T * 32 + (t + 1) * 32 + n0 + lm] = y[v];
    }
    __syncthreads();                       // protect sW/sX before next step
  }
}

// ===========================================================================
// Host-side orchestration (all on `stream`, deterministic, no allocations).
// ===========================================================================
extern "C" void kernel_launch(void* const* d_in, const int* in_sizes, int n_in,
                              void* d_out, int out_size, void* d_ws, size_t ws_size,
                              hipStream_t stream) {
  (void)n_in; (void)out_size; (void)ws_size;
  const float* x0_sys = (const float*)d_in[0];   // (B,1,32)
  const float* u_in   = (const float*)d_in[1];   // (B,T,32)
  const float* X      = (const float*)d_in[2];   // (384,384)
  const float* Y      = (const float*)d_in[3];   // (128,128)
  const float* B2     = (const float*)d_in[4];   // (128,32)
  const float* C2     = (const float*)d_in[5];   // (32,128)
  const float* D21    = (const float*)d_in[6];   // (32,128)
  const float* D22    = (const float*)d_in[7];   // (32,32)
  const float* D12    = (const float*)d_in[8];   // (128,32)
  float* out = (float*)d_out;
  float* ws  = (float*)d_ws;

  const int Bb = in_sizes[0] / 32;               // 256
  const int T  = in_sizes[1] / (Bb * 32);        // 1024

  // workspace layout (floats)
  float* H     = ws;                 // 147456
  float* E     = H + 147456;         // 16384
  float* Einv  = E + 16384;          // 16384
  float* C1w   = Einv + 16384;       // 16384
  float* Fw    = C1w + 16384;        // 16384
  float* B1w   = Fw + 16384;         // 16384
  float* Ftil  = B1w + 16384;        // 16384
  float* B1til = Ftil + 16384;       // 16384
  float* B2til = B1til + 16384;      // 4096
  float* D11p  = B2til + 4096;       // 8128
  float* Lam   = D11p + 8128;        // 128
  float* G     = Lam + 128;          // 4096
  float* x0w   = G + 4096;           // Bb*128

  const size_t INV_LDS  = (size_t)(128 * 257 + 128 + 1) * sizeof(float);
  const size_t SCAN_LDS = (size_t)SCAN_LDS_FLOATS * sizeof(float);

  // 1) H = X^T X + eps I
  k_gemm_AtA<<<dim3(24, 24), 32, 0, stream>>>(X, H);
  // 2) derive E, C1, F, B1, D11(packed), Lam
  k_derive<<<dim3(64), 256, 0, stream>>>(H, Y, E, C1w, Fw, B1w, D11p, Lam);
  // 3) E^-1
  k_invert128<<<1, 256, INV_LDS, stream>>>(E, Einv);
  // 4) fold E^-1 into weights
  k_gemm_rrr<<<dim3(8, 8), 32, 0, stream>>>(Einv, 128, Fw, 128, Ftil, 128, 128);
  k_gemm_rrr<<<dim3(8, 8), 32, 0, stream>>>(Einv, 128, B1w, 128, B1til, 128, 128);
  k_gemm_rrr<<<dim3(2, 8), 32, 0, stream>>>(Einv, 128, B2, 32, B2til, 32, 128);
  // 5) G = (C2 C2^T)^-1 C2  (pinv path)
  k_small32<<<1, 256, 0, stream>>>(C2, G);
  // 6) x0 = y0_sys @ G  (pinv(C2) applied);  y0 = x0 @ C2^T -> out[:,0,:]
  k_gemm_rrr<<<dim3(8, Bb / 16), 32, 0, stream>>>(x0_sys, 32, G, 128, x0w, 128, 32);
  k_gemm_rnt<<<dim3(2, Bb / 16), 32, 0, stream>>>(x0w, 128, C2, 128, out, T * 32, 128);
  // 7) persistent sequential scan, writes out[:,1:,:]
  k_scan<<<dim3(Bb / 16), 256, SCAN_LDS, stream>>>(
      u_in, C1w, Ftil, B1til, D11p, Lam, B2til, D12, C2, D21, D22, x0w, out, T);
}